// kernel_generated_1_21887153341272
// MI455X (gfx1250) — compile-verified
//
#include <hip/hip_runtime.h>

// MI455X / gfx1250, wave32. fp32-exact via V_WMMA_F32_16X16X4_F32.
// All zero-padding is materialized in memory/LDS so the WMMA loops are pure
// {ds_load_b64 (A), 2x global_load_b32 (B, immediate offsets), v_wmma}.

typedef float v2f __attribute__((ext_vector_type(2)));
typedef float v8f __attribute__((ext_vector_type(8)));

#define Hdim    56
#define Ssp     3136          // 56*56
#define C12     12
#define WPAD    62            // 56 + 3-halo each side
#define T3N     (C12*Hdim*WPAD)      // 41664 floats
#define CPAD    32            // t5 channel dim padded 21 -> 32
#define T5ROWS  58            // 56 + 1 zero row each side
#define T5N     (T5ROWS*Hdim*CPAD)   // 103936 floats
#define ROWBYTES1 3472        // 56*62 floats: one t3rp channel plane
#define JSTRIDE2  1792        // 56*32 floats: one spatial row of t5p

// ---------------------------------------------------------------------------
// Kernel 0: build padded+rolled channel-sum t3rp[c][n][w] (w in [0,62),
// halo w<3 / w>=59 is zero), and zero the two pad spatial rows of t5p.
// ---------------------------------------------------------------------------
__global__ void k_prep(const float* __restrict__ x,
                       float* __restrict__ t3rp, float* __restrict__ t5p) {
  int idx = blockIdx.x * blockDim.x + threadIdx.x;
  if (idx < T3N) {
    int c = idx / (Hdim * WPAD);
    int r = idx - c * (Hdim * WPAD);
    int n = r / WPAD;
    int w = r - n * WPAD;
    float v = 0.0f;
    if (w >= 3 && w < 59) {
      int m = w - 3;
      int msrc = (m + Hdim - 1) % Hdim;      // roll shift = +1
      int s = c * Ssp + n * Hdim + msrc;
      v = x[s] + x[s + C12 * Ssp];
    }
    t3rp[idx] = v;
  } else {
    int p = idx - T3N;                        // 0 .. 2*56*32-1
    if (p < 2 * Hdim * CPAD) {
      int top = p / (Hdim * CPAD);            // 0: row 0, 1: row 57
      int q = p - top * (Hdim * CPAD);
      t5p[(top ? (T5ROWS - 1) * Hdim * CPAD : 0) + q] = 0.0f;
    }
  }
}

// ---------------------------------------------------------------------------
// GEMM1: t5p[(a+1)*56+b][k] = sum_{c,jj} w2[c,jj,k] * t3rp[c][n][m+jj]
// K-order permuted: kk' = jj*12 + c  (uniform c-wrap across lanes).
// A in LDS: w2p[32][84], rows >=21 zero. One wave per 16x16 D tile.
// ---------------------------------------------------------------------------
__global__ void k_gemm1(const float* __restrict__ w2,
                        const float* __restrict__ t3rp,
                        float* __restrict__ t5p) {
  __shared__ float w2p[32 * 84];
  for (int i = threadIdx.x; i < 32 * 84; i += blockDim.x) {
    int row = i / 84;
    int kkp = i - row * 84;                   // jj*12 + c
    int jj = kkp / 12, c = kkp - jj * 12;
    w2p[i] = (row < 21) ? w2[(c * 7 + jj) * 21 + row] : 0.0f;
  }
  __syncthreads();

  int wave = (blockIdx.x * blockDim.x + threadIdx.x) >> 5;
  int lane = threadIdx.x & 31;
  int mt = wave & 1;                          // 0..1   row tile (channels)
  int nt = wave >> 1;                         // 0..195 spatial tile
  int half = lane >> 4;
  int l15  = lane & 15;

  int row = mt * 16 + l15;                    // padded channel 0..31
  int col = nt * 16 + l15;                    // spatial position
  int n = col / Hdim;
  int m = col - n * Hdim;

  const v2f*   ap = (const v2f*)&w2p[row * 84 + 2 * half];
  const float* bp = t3rp + 2 * half * ROWBYTES1 + n * WPAD + m;

  v8f acc = {};
  #pragma unroll
  for (int t = 0; t < 21; ++t) {              // K = 84
    v2f a = ap[2 * t];                        // ds_load_b64, imm offset 16t
    int off = 4 * (t % 3) * ROWBYTES1 + t / 3;   // compile-time
    v2f b;
    b.x = bp[off];                            // slot kk'
    b.y = bp[off + ROWBYTES1];                // slot kk'+1 (c+1, same jj)
    acc = __builtin_amdgcn_wmma_f32_16x16x4_f32(
        false, a, false, b, (short)0, acc, false, false);
  }

  // D: VGPR v -> M = mt*16 + v + half*8 ; N = l15.  Store shifted one spatial
  // row down into padded t5p; rows 21..31 are zeros (A was zero-padded).
  #pragma unroll
  for (int v = 0; v < 8; ++v) {
    int rout = mt * 16 + v + half * 8;        // 0..31
    t5p[(col + Hdim) * CPAD + rout] = acc[v];
  }
}

// ---------------------------------------------------------------------------
// GEMM2: y[i][a*56+b] = sum_{j,k} w1[i,j,k] * t5p[(a+j)*56+b][k]
// K padded to 96: kk' = j*32 + k (k>=21 zero).  j-wrap uniform at t=8,16.
// A in LDS: w1p[96][96].
// ---------------------------------------------------------------------------
__global__ void k_gemm2(const float* __restrict__ w1,
                        const float* __restrict__ t5p,
                        float* __restrict__ y) {
  __shared__ float w1p[96 * 96];
  for (int i = threadIdx.x; i < 96 * 96; i += blockDim.x) {
    int row = i / 96;
    int kkp = i - row * 96;                   // j*32 + k
    int j = kkp / 32, k = kkp - j * 32;
    w1p[i] = (k < 21) ? w1[row * 63 + j * 21 + k] : 0.0f;
  }
  __syncthreads();

  int wave = (blockIdx.x * blockDim.x + threadIdx.x) >> 5;
  int lane = threadIdx.x & 31;
  int mt = wave % 6;                          // 0..5   channel tile (96 = 6*16)
  int nt = wave / 6;                          // 0..195 spatial tile
  int half = lane >> 4;
  int l15  = lane & 15;

  int row = mt * 16 + l15;                    // output channel < 96
  int col = nt * 16 + l15;                    // spatial position (a*56+b)

  const v2f*   ap = (const v2f*)&w1p[row * 96 + 2 * half];
  const float* bp = t5p + col * CPAD + 2 * half;   // row a+0 in padded t5p = j=0 tap

  v8f acc = {};
  #pragma unroll
  for (int t = 0; t < 24; ++t) {              // K = 96
    v2f a = ap[2 * t];                        // ds_load_b64, imm offset 16t
    int j = t / 8;                            // compile-time
    int off = j * JSTRIDE2 + (4 * t - 32 * j);   // compile-time
    v2f b;
    b.x = bp[off];
    b.y = bp[off + 1];
    acc = __builtin_amdgcn_wmma_f32_16x16x4_f32(
        false, a, false, b, (short)0, acc, false, false);
  }

  #pragma unroll
  for (int v = 0; v < 8; ++v) {
    int rout = mt * 16 + v + half * 8;        // channel, always < 96
    y[rout * Ssp + col] = acc[v];
  }
}

// ---------------------------------------------------------------------------
extern "C" void kernel_launch(void* const* d_in, const int* in_sizes, int n_in,
                              void* d_out, int out_size, void* d_ws, size_t ws_size,
                              hipStream_t stream) {
  const float* x  = (const float*)d_in[0];    // (1,24,56,56)
  const float* w1 = (const float*)d_in[1];    // (96,3,21)
  const float* w2 = (const float*)d_in[2];    // (12,7,21)
  float* y = (float*)d_out;                   // (1,96,56,56)

  float* t3rp = (float*)d_ws;                 // 41664 floats (167 KB)
  float* t5p  = t3rp + T3N;                   // 103936 floats (416 KB)

  // Kernel 0: padded rolled channel-sum + zero t5p pad rows.
  int prep_threads = T3N + 2 * Hdim * CPAD;   // 45248
  k_prep<<<(prep_threads + 255) / 256, 256, 0, stream>>>(x, t3rp, t5p);

  // GEMM1: 2 x 196 tiles = 392 waves -> 49 blocks of 8 waves.
  k_gemm1<<<49, 256, 0, stream>>>(w2, t3rp, t5p);

  // GEMM2: 6 x 196 tiles = 1176 waves -> 147 blocks of 8 waves.
  k_gemm2<<<147, 256, 0, stream>>>(w1, t5p, y);
}